// UniformSpline_51273319580174
// MI455X (gfx1250) — compile-verified
//
#include <hip/hip_runtime.h>
#include <hip/hip_bf16.h>
#include <stdint.h>

// Problem sizes (fixed by the reference)
#define NPOS   2097152
#define NCH    8
#define NBINS  128
#define NTOT   (NPOS * NCH)   // 16,777,216 elements
#define NVEC   (NTOT / 4)     //  4,194,304 float4s

typedef float v4f __attribute__((ext_vector_type(4)));

// Table layout in d_ws (floats). Only quantities the log-det actually uses:
//   bins4 : per (ch,bin) float4 {cumheight[idx], height[idx], delta[idx], deriv[idx]}
//   edges : cumheights search array, pitch 132 (puts channels in distinct LDS banks)
//   dtab  : derivatives (for d1 = deriv[idx+1]), pitch 132
#define BINS4_OFF  0                    // 8*128 float4  = 4096 floats
#define EDGES_OFF  4096                 // 8*132         = 1056 floats
#define DTAB_OFF   5152                 // 8*132         = 1056 floats
#define TBL_FLOATS 6208
#define TBL_BYTES  (TBL_FLOATS * 4)     // 24,832 B
#define TBL_CHUNKS (TBL_BYTES / 16)     // 1552 b128 chunks (exact)

// ---------------------------------------------------------------------------
// Kernel 1: build spline tables. 8 threads, one per channel; serial per
// channel (8 x ~500 flops total — launch latency dominates, keep it simple
// and faithful to the reference math).
// ---------------------------------------------------------------------------
__global__ void build_tables_kernel(const float* __restrict__ uw,
                                    const float* __restrict__ uh,
                                    const float* __restrict__ ud,
                                    float* __restrict__ tbl) {
  const int c = threadIdx.x;
  if (c >= NCH) return;

  const float MIN_SZ  = 1e-3f;               // MIN_BIN_WIDTH == MIN_BIN_HEIGHT
  const float SPAN_SC = 1.0f - 128.0f * MIN_SZ;

  // softmax max-subtraction (matches jax.nn.softmax numerics)
  float mw = -1e30f, mh = -1e30f;
  for (int b = 0; b < NBINS; ++b) {
    mw = fmaxf(mw, uw[c * NBINS + b] * 10.0f);
    mh = fmaxf(mh, uh[c * NBINS + b] * 10.0f);
  }
  float sw = 0.0f, sh = 0.0f;
  for (int b = 0; b < NBINS; ++b) {
    sw += __expf(uw[c * NBINS + b] * 10.0f - mw);
    sh += __expf(uh[c * NBINS + b] * 10.0f - mh);
  }
  const float rw = 1.0f / sw, rh = 1.0f / sh;

  v4f*   bins4 = (v4f*)(tbl + BINS4_OFF);
  float* edges = tbl + EDGES_OFF;
  float* dtab  = tbl + DTAB_OFF;

  // derivatives: boundary pads use DEFAULT_INIT; softplus(DEFAULT_INIT) is
  // exactly 1 - MIN_DERIVATIVE, so boundary derivative == 1.0
  dtab[c * 132 + 0]   = 1.0f;
  dtab[c * 132 + 128] = 1.0f;
  for (int k = 1; k <= 127; ++k) {
    float u = ud[c * 127 + (k - 1)];
    dtab[c * 132 + k] = 1e-3f + log1pf(__expf(u));   // MIN_DERIVATIVE + softplus
  }
  dtab[c * 132 + 129] = dtab[c * 132 + 130] = dtab[c * 132 + 131] = 1.0f;

  // streaming cumsum for widths/heights; endpoints clamped to exactly +-1
  float fw = 0.0f, fh = 0.0f;
  float cwp = -1.0f, chp = -1.0f;
  edges[c * 132 + 0] = -1.0f;
  for (int b = 0; b < NBINS; ++b) {
    float pw = MIN_SZ + SPAN_SC * (__expf(uw[c * NBINS + b] * 10.0f - mw) * rw);
    float ph = MIN_SZ + SPAN_SC * (__expf(uh[c * NBINS + b] * 10.0f - mh) * rh);
    fw += pw;
    fh += ph;
    float cwn = (b == NBINS - 1) ? 1.0f : (2.0f * fw - 1.0f);
    float chn = (b == NBINS - 1) ? 1.0f : (2.0f * fh - 1.0f);
    float w = cwn - cwp;
    float h = chn - chp;
    v4f rec;
    rec.x = chp; rec.y = h; rec.z = h / w; rec.w = dtab[c * 132 + b];
    bins4[(c << 7) + b] = rec;
    // search edges use cumheights with last element bumped by 1e-6
    edges[c * 132 + b + 1] = (b == NBINS - 1) ? (1.0f + 1e-6f) : chn;
    cwp = cwn;
    chp = chn;
  }
  edges[c * 132 + 129] = edges[c * 132 + 130] = edges[c * 132 + 131] = 3.0f;
}

// ---------------------------------------------------------------------------
// Kernel 2: streaming log-det evaluation. Tables staged into LDS via CDNA5
// async global->LDS DMA; x/out streamed non-temporally (once-through 128MB).
// ---------------------------------------------------------------------------
__global__ __launch_bounds__(256) void spline_logprob_kernel(
    const float* __restrict__ x,
    const float* __restrict__ tbl,
    float* __restrict__ out) {
  __shared__ alignas(16) float smem[TBL_FLOATS];

  const int tid = threadIdx.x;

  // Low 32 bits of a flat LDS pointer == byte offset within the wave's LDS
  // allocation (aperture mapping). Taking ptrtoint of smem also *captures* it,
  // so the "memory"-clobbering asm below is a legal writer of smem — without
  // this, LLVM proves smem is never stored and folds every LDS read to undef.
  const unsigned lds_base = (unsigned)(uintptr_t)&smem[0];

  // CDNA5 async copy: 1552 x b128 chunks, tracked on ASYNCcnt
  for (int i = tid; i < TBL_CHUNKS; i += 256) {
    unsigned           lofs = lds_base + (unsigned)(i * 16);
    unsigned long long gadr = (unsigned long long)(uintptr_t)tbl
                            + (unsigned long long)i * 16ull;
    asm volatile("global_load_async_to_lds_b128 %0, %1, off"
                 :: "v"(lofs), "v"(gadr), "s"(&smem[0]) : "memory");
  }
  asm volatile("s_wait_asynccnt 0x0" ::: "memory");
  __syncthreads();

  const v4f*   bins4 = (const v4f*)(smem + BINS4_OFF);
  const float* edges = smem + EDGES_OFF;
  const float* dtab  = smem + DTAB_OFF;

  const v4f* xv4 = (const v4f*)x;
  v4f*       ov4 = (v4f*)out;
  const unsigned stride = gridDim.x * blockDim.x;

  for (unsigned v = blockIdx.x * blockDim.x + tid; v < NVEC; v += stride) {
    v4f xv = __builtin_nontemporal_load(xv4 + v);
    const int c0 = (int)((v * 4u) & 7u);   // float4 covers channels {0..3} or {4..7}
    float xin[4] = {xv.x, xv.y, xv.z, xv.w};
    float r[4];

#pragma unroll
    for (int j = 0; j < 4; ++j) {
      const int c  = c0 + j;
      float     xx = xin[j];
      bool inside  = (xx >= -1.0f) && (xx <= 1.0f);
      float vv     = fminf(fmaxf(xx, -1.0f), 1.0f);

      // branchless binary search: largest idx in [0,127] with edges[idx] <= vv
      const float* e = edges + c * 132;
      int idx = 0;
#pragma unroll
      for (int s = 64; s >= 1; s >>= 1) {
        int p = idx + s;
        idx = (e[p] <= vv) ? p : idx;
      }

      v4f bb     = bins4[(c << 7) + idx];        // one ds_load_b128
      float ich  = bb.x, ih = bb.y, idl = bb.z, d0 = bb.w;
      float d1   = dtab[c * 132 + idx + 1];      // one ds_load_b32

      // quadratic inverse of the rational-quadratic spline
      float dy   = vv - ich;
      float t    = d0 + d1 - 2.0f * idl;
      float aa   = dy * t + ih * (idl - d0);
      float b2   = ih * d0 - dy * t;
      float cc   = -idl * dy;
      float disc = fmaxf(b2 * b2 - 4.0f * aa * cc, 0.0f);
      float root = (2.0f * cc) *
                   __builtin_amdgcn_rcpf(-b2 - __builtin_amdgcn_sqrtf(disc));
      float th1m = root * (1.0f - root);
      float omr  = 1.0f - root;
      float den  = idl + t * th1m;
      float dnum = idl * idl * (d1 * root * root + 2.0f * idl * th1m + d0 * omr * omr);
      float lad  = __logf(dnum) - 2.0f * __logf(den);

      r[j] = -0.69314718056f - (inside ? lad : 0.0f);   // log(0.5) - logabsdet
    }

    v4f ov;
    ov.x = r[0]; ov.y = r[1]; ov.z = r[2]; ov.w = r[3];
    __builtin_nontemporal_store(ov, ov4 + v);
  }
}

// ---------------------------------------------------------------------------
extern "C" void kernel_launch(void* const* d_in, const int* in_sizes, int n_in,
                              void* d_out, int out_size, void* d_ws, size_t ws_size,
                              hipStream_t stream) {
  (void)in_sizes; (void)n_in; (void)out_size; (void)ws_size;
  const float* x  = (const float*)d_in[0];
  const float* uw = (const float*)d_in[1];
  const float* uh = (const float*)d_in[2];
  const float* ud = (const float*)d_in[3];
  float* out = (float*)d_out;
  float* tbl = (float*)d_ws;   // 24,832 bytes of scratch used

  build_tables_kernel<<<1, 8, 0, stream>>>(uw, uh, ud, tbl);
  spline_logprob_kernel<<<2048, 256, 0, stream>>>(x, tbl, out);
}